// PointTransformer_90323162235707
// MI455X (gfx1250) — compile-verified
//
#include <hip/hip_runtime.h>
#include <math.h>

// ---------------------------------------------------------------------------
// Types for CDNA5 WMMA f32 16x16x4:  A = 16x4 (2 VGPR), B = 4x16 (2 VGPR),
// C/D = 16x16 f32 (8 VGPR).
// ---------------------------------------------------------------------------
typedef float v2f __attribute__((ext_vector_type(2)));
typedef float v8f __attribute__((ext_vector_type(8)));

#define DM 512   // D_MODEL
#define KNN 16   // NNEIGHBOR
#define BATCH 8

static __device__ __forceinline__ int imin(int a, int b) { return a < b ? a : b; }

// ---------------------------------------------------------------------------
// Generic GEMM:  Y[M,N] = act( scale[n] * (X[M,K] @ W[N,K]^T) + shift[n] ) (+res)
// One 32x32 region per wave (2x2 WMMA tiles), 4 waves / 128-thread block.
// Edge handling:
//   - M/N edges: row indices CLAMPED (always-valid loads, no branches);
//     garbage only reaches out-of-range output rows/cols which are not stored.
//   - K tail: value-masked (multiply by 0/1, v_cndmask — branchless).
// EXEC is all-1 at every WMMA (no divergent control flow around them).
// ---------------------------------------------------------------------------
__global__ void gemm_wmma_f32(const float* __restrict__ X,
                              const float* __restrict__ W,
                              const float* __restrict__ scale,
                              const float* __restrict__ shift,
                              const float* __restrict__ residual,
                              float* __restrict__ Y,
                              int M, int Nc, int K, int relu)
{
    const int lane = threadIdx.x & 31;
    const int wave = threadIdx.x >> 5;
    const int n0 = blockIdx.x * 32;
    const int m0 = (blockIdx.y * 4 + wave) * 32;
    if (m0 >= M) return;                    // wave-uniform exit (EXEC stays full)

    const int half = lane >> 4;             // 0: lanes 0-15, 1: lanes 16-31
    const int lid  = lane & 15;

    // Clamped row pointers: loads are always in-bounds; clamped (duplicate)
    // rows only influence out-of-range outputs which are never stored.
    const float* __restrict__ xp0 = X + (size_t)imin(m0 + lid,      M - 1) * K;
    const float* __restrict__ xp1 = X + (size_t)imin(m0 + 16 + lid, M - 1) * K;
    const float* __restrict__ wp0 = W + (size_t)imin(n0 + lid,      Nc - 1) * K;
    const float* __restrict__ wp1 = W + (size_t)imin(n0 + 16 + lid, Nc - 1) * K;

    v8f acc00 = {0.f,0.f,0.f,0.f,0.f,0.f,0.f,0.f};
    v8f acc01 = acc00, acc10 = acc00, acc11 = acc00;

    const int K4 = K & ~3;
    for (int k0 = 0; k0 < K4; k0 += 4) {
        const int ka = k0 + 2 * half;       // K pair handled by this half-wave
        v2f a0, a1, b0, b1;
        a0.x = xp0[ka]; a0.y = xp0[ka + 1];
        a1.x = xp1[ka]; a1.y = xp1[ka + 1];
        b0.x = wp0[ka]; b0.y = wp0[ka + 1];
        b1.x = wp1[ka]; b1.y = wp1[ka + 1];
        acc00 = __builtin_amdgcn_wmma_f32_16x16x4_f32(false, a0, false, b0, (short)0, acc00, false, false);
        acc01 = __builtin_amdgcn_wmma_f32_16x16x4_f32(false, a0, false, b1, (short)0, acc01, false, false);
        acc10 = __builtin_amdgcn_wmma_f32_16x16x4_f32(false, a1, false, b0, (short)0, acc10, false, false);
        acc11 = __builtin_amdgcn_wmma_f32_16x16x4_f32(false, a1, false, b1, (short)0, acc11, false, false);
    }
    if (K4 < K) {                           // branchless K tail (runs once)
        const int ka = K4 + 2 * half;
        const int i0 = imin(ka,     K - 1);
        const int i1 = imin(ka + 1, K - 1);
        const float mk0 = (ka     < K) ? 1.f : 0.f;
        const float mk1 = (ka + 1 < K) ? 1.f : 0.f;
        v2f a0, a1, b0, b1;
        a0.x = xp0[i0] * mk0; a0.y = xp0[i1] * mk1;
        a1.x = xp1[i0] * mk0; a1.y = xp1[i1] * mk1;
        b0.x = wp0[i0] * mk0; b0.y = wp0[i1] * mk1;
        b1.x = wp1[i0] * mk0; b1.y = wp1[i1] * mk1;
        acc00 = __builtin_amdgcn_wmma_f32_16x16x4_f32(false, a0, false, b0, (short)0, acc00, false, false);
        acc01 = __builtin_amdgcn_wmma_f32_16x16x4_f32(false, a0, false, b1, (short)0, acc01, false, false);
        acc10 = __builtin_amdgcn_wmma_f32_16x16x4_f32(false, a1, false, b0, (short)0, acc10, false, false);
        acc11 = __builtin_amdgcn_wmma_f32_16x16x4_f32(false, a1, false, b1, (short)0, acc11, false, false);
    }

    // Epilogue: 2x2 tiles; tile (tm,tn) row = m0+16*tm + r + 8*half, col = n0+16*tn+lid
    const v8f* accs[2][2] = { {&acc00, &acc01}, {&acc10, &acc11} };
#pragma unroll
    for (int tn = 0; tn < 2; ++tn) {
        const int col = n0 + 16 * tn + lid;
        if (col >= Nc) continue;
        const float sc = scale ? scale[col] : 1.f;
        const float sh = shift ? shift[col] : 0.f;
#pragma unroll
        for (int tm = 0; tm < 2; ++tm) {
            const v8f a = *accs[tm][tn];
#pragma unroll
            for (int r = 0; r < 8; ++r) {
                const int row = m0 + 16 * tm + r + 8 * half;
                if (row < M) {
                    float v = a[r] * sc + sh;
                    if (residual) v += residual[(size_t)row * Nc + col];
                    if (relu && v < 0.f) v = 0.f;
                    Y[(size_t)row * Nc + col] = v;
                }
            }
        }
    }
}

// ---------------------------------------------------------------------------
// kNN (k=16 smallest squared distances, stable tie order = ascending index).
// q: (B,Mq,3), ref: (B,Nr,3), idx: (B,Mq,16). blockIdx.y = batch.
// ---------------------------------------------------------------------------
__global__ void knn_kernel(const float* __restrict__ q,
                           const float* __restrict__ ref,
                           int* __restrict__ idx, int Mq, int Nr)
{
    const int m = blockIdx.x * blockDim.x + threadIdx.x;
    const int b = blockIdx.y;
    if (m >= Mq) return;
    const float* qp = q + ((size_t)b * Mq + m) * 3;
    const float qx = qp[0], qy = qp[1], qz = qp[2];
    float best[KNN]; int bidx[KNN];
#pragma unroll
    for (int j = 0; j < KNN; ++j) { best[j] = 3.4e38f; bidx[j] = 0; }
    const float* rp = ref + (size_t)b * Nr * 3;
    for (int n = 0; n < Nr; ++n) {
        const float dx = qx - rp[3*n], dy = qy - rp[3*n+1], dz = qz - rp[3*n+2];
        const float d = dx*dx + dy*dy + dz*dz;
        if (d < best[KNN-1]) {
            int j = KNN - 1;
            while (j > 0 && d < best[j-1]) { best[j] = best[j-1]; bidx[j] = bidx[j-1]; --j; }
            best[j] = d; bidx[j] = n;
        }
    }
    int* op = idx + ((size_t)b * Mq + m) * KNN;
#pragma unroll
    for (int j = 0; j < KNN; ++j) op[j] = bidx[j];
}

// ---------------------------------------------------------------------------
// Farthest point sampling. One block (256 threads) per batch. N <= 1024.
// ---------------------------------------------------------------------------
__global__ void fps_kernel(const float* __restrict__ xyz, int* __restrict__ fidx,
                           int N, int npoint)
{
    __shared__ float dist[1024];
    __shared__ float sval[256];
    __shared__ int   sidx[256];
    __shared__ int   far;
    const int b = blockIdx.x;
    const int t = threadIdx.x;
    const float* xp = xyz + (size_t)b * N * 3;
    for (int i = t; i < N; i += 256) dist[i] = 1e10f;
    if (t == 0) far = 0;
    __syncthreads();
    for (int it = 0; it < npoint; ++it) {
        const int f = far;
        if (t == 0) fidx[(size_t)b * npoint + it] = f;
        const float cx = xp[3*f], cy = xp[3*f+1], cz = xp[3*f+2];
        float lbv = -1.f; int lbi = 0;
        for (int i = t; i < N; i += 256) {
            const float dx = xp[3*i]-cx, dy = xp[3*i+1]-cy, dz = xp[3*i+2]-cz;
            float dm = dx*dx + dy*dy + dz*dz;
            const float old = dist[i];
            if (old < dm) dm = old;
            dist[i] = dm;
            if (dm > lbv) { lbv = dm; lbi = i; }
        }
        sval[t] = lbv; sidx[t] = lbi;
        __syncthreads();
        for (int s = 128; s > 0; s >>= 1) {
            if (t < s) {
                if (sval[t+s] > sval[t] ||
                    (sval[t+s] == sval[t] && sidx[t+s] < sidx[t])) {
                    sval[t] = sval[t+s]; sidx[t] = sidx[t+s];
                }
            }
            __syncthreads();
        }
        if (t == 0) far = sidx[0];
        __syncthreads();
    }
}

// new_xyz[b,m,:] = xyz[b, fidx[b,m], :]   (blockIdx.y = batch)
__global__ void gather_xyz(const float* __restrict__ xyz, const int* __restrict__ fidx,
                           float* __restrict__ out, int N, int npoint)
{
    const int t = blockIdx.x * blockDim.x + threadIdx.x;
    const int b = blockIdx.y;
    if (t >= npoint) return;
    const int id = fidx[(size_t)b * npoint + t];
    const float* s = xyz + ((size_t)b * N + id) * 3;
    float* o = out + ((size_t)b * npoint + t) * 3;
    o[0] = s[0]; o[1] = s[1]; o[2] = s[2];
}

// delta[b,m,j,:] = qxyz[b,m,:] - rxyz[b, idx[b,m,j], :]  (query - neighbor)
__global__ void build_delta(const float* __restrict__ qxyz, const float* __restrict__ rxyz,
                            const int* __restrict__ idx, float* __restrict__ out,
                            int Mq, int Nr)
{
    const int t = blockIdx.x * blockDim.x + threadIdx.x;
    if (t >= Mq * KNN) return;
    const int b = blockIdx.y;
    const int m = t >> 4;
    const int id = idx[(size_t)b * Mq * KNN + t];
    const float* qp = qxyz + ((size_t)b * Mq + m) * 3;
    const float* rp = rxyz + ((size_t)b * Nr + id) * 3;
    float* op = out + ((size_t)b * Mq * KNN + t) * 3;
    op[0] = qp[0] - rp[0];
    op[1] = qp[1] - rp[1];
    op[2] = qp[2] - rp[2];
}

// Per-batch neighbor-tensor builder (D_MODEL=512), one block per (n,j):
//   mode 0: out[n,j,f] = a[n,f] - g[idx[n,j],f] + pos[n,j,f]   (q - k + pos)
//   mode 1: out[n,j,f] =          g[idx[n,j],f] + pos[n,j,f]   (v + pos)
__global__ void build_nbr(const float* __restrict__ a, const float* __restrict__ g,
                          const float* __restrict__ pos, const int* __restrict__ idx,
                          float* __restrict__ out, int mode)
{
    const int nj = blockIdx.x;
    const int n  = nj >> 4;
    const int id = idx[nj];
    const float* gp = g + (size_t)id * DM;
    const float* pp = pos + (size_t)nj * DM;
    float* op = out + (size_t)nj * DM;
    const float* ap = a + (size_t)n * DM;   // unused when mode==1
    for (int f = threadIdx.x; f < DM; f += blockDim.x) {
        float v = pp[f] + gp[f] * (mode ? 1.f : -1.f);
        if (mode == 0) v += ap[f];
        op[f] = v;
    }
}

// softmax over the 16 neighbors (axis=-2), per (n,f); attn (Npts,16,512), in place.
__global__ void softmax_k(float* __restrict__ attn, int Npts)
{
    const long t = (long)blockIdx.x * blockDim.x + threadIdx.x;
    if (t >= (long)Npts * DM) return;
    const int n = (int)(t >> 9);
    const int f = (int)(t & 511);
    float* p = attn + ((long)n << 13) + f;      // n*16*512
    const float sc = 0.04419417382415922f;      // 1/sqrt(512)
    float mx = -3.4e38f;
#pragma unroll
    for (int j = 0; j < KNN; ++j) { const float v = p[(long)j << 9] * sc; if (v > mx) mx = v; }
    float e[KNN]; float sum = 0.f;
#pragma unroll
    for (int j = 0; j < KNN; ++j) { e[j] = __expf(p[(long)j << 9] * sc - mx); sum += e[j]; }
    const float inv = 1.f / sum;
#pragma unroll
    for (int j = 0; j < KNN; ++j) p[(long)j << 9] = e[j] * inv;
}

// res[n,f] = sum_j attn[n,j,f] * vpos[n,j,f]
__global__ void attn_sum(const float* __restrict__ attn, const float* __restrict__ vpos,
                         float* __restrict__ res, int Npts)
{
    const long t = (long)blockIdx.x * blockDim.x + threadIdx.x;
    if (t >= (long)Npts * DM) return;
    const int n = (int)(t >> 9);
    const int f = (int)(t & 511);
    const float* ap = attn + ((long)n << 13) + f;
    const float* vp = vpos + ((long)n << 13) + f;
    float s = 0.f;
#pragma unroll
    for (int j = 0; j < KNN; ++j) s += ap[(long)j << 9] * vp[(long)j << 9];
    res[t] = s;
}

// xcat[b,m,j,:] = [ xyz[b,idx]-new_xyz[b,m] (3) | points[b,idx] (Cf) ]
__global__ void build_xcat(const float* __restrict__ xyz, const float* __restrict__ nxyz,
                           const float* __restrict__ pts, const int* __restrict__ idx,
                           float* __restrict__ out, int npoint, int N, int Cf)
{
    const int b = blockIdx.y;
    const int mj = blockIdx.x;                   // npoint*16
    const int m = mj >> 4;
    const int id = idx[(size_t)b * npoint * KNN + mj];
    const int inc = Cf + 3;
    float* op = out + ((size_t)b * npoint * KNN + mj) * (size_t)inc;
    const float* xp = xyz  + ((size_t)b * N + id) * 3;
    const float* cp = nxyz + ((size_t)b * npoint + m) * 3;
    const float* pp = pts  + ((size_t)b * N + id) * (size_t)Cf;
    for (int c = threadIdx.x; c < inc; c += blockDim.x)
        op[c] = (c < 3) ? (xp[c] - cp[c]) : pp[c - 3];
}

// out[r,f] = max_j x[r,j,f], j in [0,16)
__global__ void max_over_k(const float* __restrict__ x, float* __restrict__ out,
                           long rows, int C)
{
    const long t = (long)blockIdx.x * blockDim.x + threadIdx.x;
    if (t >= rows * (long)C) return;
    const long r = t / C; const int f = (int)(t % C);
    const float* p = x + (r * KNN) * (long)C + f;
    float m = p[0];
#pragma unroll
    for (int j = 1; j < KNN; ++j) { const float v = p[(long)j * C]; if (v > m) m = v; }
    out[t] = m;
}

// out[b,f] = max_n x[b,n,f]
__global__ void max_axis1(const float* __restrict__ x, float* __restrict__ out,
                          int Bv, int N, int C)
{
    const int t = blockIdx.x * blockDim.x + threadIdx.x;
    if (t >= Bv * C) return;
    const int b = t / C, f = t % C;
    const float* p = x + ((size_t)b * N) * C + f;
    float m = p[0];
    for (int n = 1; n < N; ++n) { const float v = p[(size_t)n * C]; if (v > m) m = v; }
    out[t] = m;
}

// Fold BatchNorm into per-channel affine: scale = gm*rsqrt(rv+eps),
// shift = scale*(bias - rm) + bt     (y = relu(scale * (x@W^T) + shift))
__global__ void bn_fold(const float* __restrict__ bias, const float* __restrict__ gm,
                        const float* __restrict__ bt, const float* __restrict__ rm,
                        const float* __restrict__ rv, float* __restrict__ scale,
                        float* __restrict__ shift, int C)
{
    const int i = blockIdx.x * blockDim.x + threadIdx.x;
    if (i >= C) return;
    const float s = gm[i] * rsqrtf(rv[i] + 1e-5f);
    scale[i] = s;
    shift[i] = s * (bias[i] - rm[i]) + bt[i];
}

// feat[b, 0:512]=maxfeat, feat[b,512:520]=meta
__global__ void concat_feat(const float* __restrict__ mx, const float* __restrict__ meta,
                            float* __restrict__ out)
{
    const int t = blockIdx.x * blockDim.x + threadIdx.x;
    if (t >= BATCH * 520) return;
    const int b = t / 520, i = t % 520;
    out[t] = (i < 512) ? mx[b * 512 + i] : meta[b * 8 + (i - 512)];
}

// d_out[j*8+b] = pred[b*3+j]  (tuple of 3 length-8 arrays, concatenated)
__global__ void transpose_out(const float* __restrict__ pred, float* __restrict__ out)
{
    const int t = threadIdx.x;
    if (t < 24) { const int b = t / 3, j = t % 3; out[j * 8 + b] = pred[t]; }
}

// ---------------------------------------------------------------------------
// Host orchestration
// ---------------------------------------------------------------------------
extern "C" void kernel_launch(void* const* d_in, const int* in_sizes, int n_in,
                              void* d_out, int out_size, void* d_ws, size_t ws_size,
                              hipStream_t stream)
{
    (void)in_sizes; (void)n_in; (void)out_size; (void)ws_size;
    auto P = [&](int i) -> const float* { return (const float*)d_in[i]; };

    // ---- workspace carving (deterministic, no allocation) -----------------
    char* base = (char*)d_ws;
    size_t off = 0;
    auto alloc = [&](size_t nbytes) -> void* {
        void* p = base + off;
        off += (nbytes + 255) & ~(size_t)255;
        return p;
    };
    float* FH   = (float*)alloc(sizeof(float) * 8ull * 1024 * 32);     // fc1 hidden
    float* fA   = (float*)alloc(sizeof(float) * 8ull * 1024 * 32);     // feats ping
    float* fB   = (float*)alloc(sizeof(float) * 8ull * 1024 * 32);     // feats pong
    float* X512 = (float*)alloc(sizeof(float) * 8ull * 1024 * DM);     // tf fc1 out
    float* QB   = (float*)alloc(sizeof(float) * 8ull * 1024 * DM);
    float* KB   = (float*)alloc(sizeof(float) * 8ull * 1024 * DM);
    float* VB   = (float*)alloc(sizeof(float) * 8ull * 1024 * DM);
    float* RES  = (float*)alloc(sizeof(float) * 8ull * 1024 * DM);
    float* POS  = (float*)alloc(sizeof(float) * 1024ull * KNN * DM);   // per-batch
    float* BUFA = (float*)alloc(sizeof(float) * 1024ull * KNN * DM);   // per-batch
    float* BUFC = (float*)alloc(sizeof(float) * 1024ull * KNN * DM);   // per-batch
    float* DELTA= (float*)alloc(sizeof(float) * 8ull * 1024 * KNN * 3);
    int*   IDX  = (int*)  alloc(sizeof(int)   * 8ull * 1024 * KNN);
    int*   FIDX = (int*)  alloc(sizeof(int)   * 8ull * 256);
    float* XYZA = (float*)alloc(sizeof(float) * 8ull * 256 * 3);
    float* XYZB = (float*)alloc(sizeof(float) * 8ull * 256 * 3);
    float* XCAT = (float*)alloc(sizeof(float) * 8ull * 256 * KNN * 35); // max level
    float* TDH1 = (float*)alloc(sizeof(float) * 8ull * 256 * KNN * 64); // max level
    float* TDH2 = (float*)alloc(sizeof(float) * 8ull * 256 * KNN * 64);
    float* BNS  = (float*)alloc(sizeof(float) * 512);
    float* BNH  = (float*)alloc(sizeof(float) * 512);
    float* MAXF = (float*)alloc(sizeof(float) * 8 * 512);
    float* MH   = (float*)alloc(sizeof(float) * 8 * 16);
    float* META = (float*)alloc(sizeof(float) * 8 * 8);
    float* FEAT = (float*)alloc(sizeof(float) * 8 * 520);
    float* H1   = (float*)alloc(sizeof(float) * 8 * 256);
    float* H2   = (float*)alloc(sizeof(float) * 8 * 64);
    float* PRED = (float*)alloc(sizeof(float) * 32);

    auto gemm = [&](const float* X, const float* W, const float* sc, const float* sh,
                    const float* res, float* Y, int M, int N, int K, int relu) {
        dim3 g((N + 31) / 32, (M + 127) / 128);   // 4 waves x 32 rows per block
        gemm_wmma_f32<<<g, 128, 0, stream>>>(X, W, sc, sh, res, Y, M, N, K, relu);
    };

    // ---- transformer block (params at pb..pb+14) --------------------------
    auto tf_block = [&](int pb, const float* xyz, float* featsIn, float* featsOut,
                        int Npts, int dp) {
        { dim3 g((Npts + 63) / 64, BATCH);
          knn_kernel<<<g, 64, 0, stream>>>(xyz, xyz, IDX, Npts, Npts); }
        const int M = BATCH * Npts;
        gemm(featsIn, P(pb + 0), nullptr, P(pb + 1), nullptr, X512, M, DM, dp, 0); // fc1
        gemm(X512, P(pb + 12), nullptr, nullptr, nullptr, QB, M, DM, DM, 0);       // wq
        gemm(X512, P(pb + 13), nullptr, nullptr, nullptr, KB, M, DM, DM, 0);       // wk
        gemm(X512, P(pb + 14), nullptr, nullptr, nullptr, VB, M, DM, DM, 0);       // wv
        { dim3 g((Npts * KNN + 127) / 128, BATCH);
          build_delta<<<g, 128, 0, stream>>>(xyz, xyz, IDX, DELTA, Npts, Npts); }
        const int Mnk = Npts * KNN;
        const long tot = (long)Npts * DM;
        for (int b = 0; b < BATCH; ++b) {
            const float* dlt = DELTA + (size_t)b * Mnk * 3;
            const int* idb = IDX + (size_t)b * Mnk;
            gemm(dlt, P(pb + 4), nullptr, P(pb + 5), nullptr, BUFA, Mnk, DM, 3, 1);   // fc_delta0+relu
            gemm(BUFA, P(pb + 6), nullptr, P(pb + 7), nullptr, POS, Mnk, DM, DM, 0);  // fc_delta1
            build_nbr<<<Mnk, 256, 0, stream>>>(QB + (size_t)b * Npts * DM,
                                               KB + (size_t)b * Npts * DM,
                                               POS, idb, BUFA, 0);                     // q-k+pos
            gemm(BUFA, P(pb + 8), nullptr, P(pb + 9), nullptr, BUFC, Mnk, DM, DM, 1);  // fc_gamma0+relu
            gemm(BUFC, P(pb + 10), nullptr, P(pb + 11), nullptr, BUFA, Mnk, DM, DM, 0);// fc_gamma1
            softmax_k<<<(int)((tot + 255) / 256), 256, 0, stream>>>(BUFA, Npts);
            build_nbr<<<Mnk, 256, 0, stream>>>(nullptr,
                                               VB + (size_t)b * Npts * DM,
                                               POS, idb, BUFC, 1);                     // v+pos
            attn_sum<<<(int)((tot + 255) / 256), 256, 0, stream>>>(
                BUFA, BUFC, RES + (size_t)b * Npts * DM, Npts);
        }
        gemm(RES, P(pb + 2), nullptr, P(pb + 3), featsIn, featsOut, M, dp, DM, 0);     // fc2 + pre
    };

    // ---- transition down (params at pb..pb+11) ----------------------------
    auto td_block = [&](int pb, const float* xyz, const float* featsIn, float* xyzOut,
                        float* featsOut, int N, int np_out, int c) {
        const int Cf = c / 2, inc = Cf + 3;
        fps_kernel<<<BATCH, 256, 0, stream>>>(xyz, FIDX, N, np_out);
        { dim3 g((np_out + 63) / 64, BATCH);
          gather_xyz<<<g, 64, 0, stream>>>(xyz, FIDX, xyzOut, N, np_out); }
        { dim3 g((np_out + 63) / 64, BATCH);
          knn_kernel<<<g, 64, 0, stream>>>(xyzOut, xyz, IDX, np_out, N); }
        { dim3 g(np_out * KNN, BATCH);
          build_xcat<<<g, 64, 0, stream>>>(xyz, xyzOut, featsIn, IDX, XCAT, np_out, N, Cf); }
        const int M = BATCH * np_out * KNN;
        bn_fold<<<(c + 63) / 64, 64, 0, stream>>>(P(pb+1), P(pb+2), P(pb+3), P(pb+4), P(pb+5), BNS, BNH, c);
        gemm(XCAT, P(pb + 0), BNS, BNH, nullptr, TDH1, M, c, inc, 1);
        bn_fold<<<(c + 63) / 64, 64, 0, stream>>>(P(pb+7), P(pb+8), P(pb+9), P(pb+10), P(pb+11), BNS, BNH, c);
        gemm(TDH1, P(pb + 6), BNS, BNH, nullptr, TDH2, M, c, c, 1);
        { long rows = (long)BATCH * np_out; long t2 = rows * c;
          max_over_k<<<(int)((t2 + 255) / 256), 256, 0, stream>>>(TDH2, featsOut, rows, c); }
    };

    // ---- forward pass -----------------------------------------------------
    const float* lidar = (const float*)d_in[0];   // (8,1024,3); xyz == lidar
    // fc1: 3 -> 32 relu -> 32
    gemm(lidar, P(2), nullptr, P(3), nullptr, FH, BATCH * 1024, 32, 3, 1);
    gemm(FH, P(4), nullptr, P(5), nullptr, fA, BATCH * 1024, 32, 32, 0);
    // t1 (params 16..30)
    tf_block(16, lidar, fA, fB, 1024, 32);
    { float* t = fA; fA = fB; fB = t; }

    const float* cxyz = lidar;
    float* xA = XYZA; float* xB = XYZB;
    int Npts = 1024;
    for (int i = 0; i < 4; ++i) {
        const int np_out = Npts / 4;
        const int c = 32 << (i + 1);              // 64,128,256,512
        td_block(31 + 12 * i, cxyz, fA, xA, fB, Npts, np_out, c);
        cxyz = xA; { float* t = xA; xA = xB; xB = t; }
        { float* t = fA; fA = fB; fB = t; }
        tf_block(79 + 15 * i, cxyz, fA, fB, np_out, c);
        { float* t = fA; fA = fB; fB = t; }
        Npts = np_out;
    }

    // final head: points (8,4,512) in fA
    max_axis1<<<(BATCH * 512 + 255) / 256, 256, 0, stream>>>(fA, MAXF, BATCH, Npts, 512);
    gemm((const float*)d_in[1], P(12), nullptr, P(13), nullptr, MH, BATCH, 16, 1, 1); // fc_meta0+relu
    gemm(MH, P(14), nullptr, P(15), nullptr, META, BATCH, 8, 16, 0);                  // fc_meta1
    concat_feat<<<(BATCH * 520 + 255) / 256, 256, 0, stream>>>(MAXF, META, FEAT);
    gemm(FEAT, P(6), nullptr, P(7), nullptr, H1, BATCH, 256, 520, 1);
    gemm(H1, P(8), nullptr, P(9), nullptr, H2, BATCH, 64, 256, 1);
    gemm(H2, P(10), nullptr, P(11), nullptr, PRED, BATCH, 3, 64, 0);
    transpose_out<<<1, 32, 0, stream>>>(PRED, (float*)d_out);
}